// TreeTrackProcesser_25769803776218
// MI455X (gfx1250) — compile-verified
//
#include <hip/hip_runtime.h>
#include <hip/hip_bf16.h>

// ---------------------------------------------------------------------------
// TreeTrackProcesser: GCN-style track conv on MI455X (gfx1250, wave32)
//   - deg/dinv/norm build (hardware f32 atomics)
//   - input layer GEMM via V_WMMA_F32_16X16X4_F32 (f32 tensor path)
//   - 8 scatter SpMMs, wave-per-edge, float4 gathers, global_atomic_add_f32
//   - final attention + head contraction
// Entire working set (~60 MB) lives in the 192 MB L2.
// ---------------------------------------------------------------------------

typedef float v2f __attribute__((ext_vector_type(2)));
typedef float v8f __attribute__((ext_vector_type(8)));

#define T_HIDDEN 32
#define T_D      64      // HIDDEN * N_HEADS
#define T_CD     128     // N_CLASSES * D
#define T_FW     0.5f    // FEAT_W

__device__ __forceinline__ void atomAddF(float* p, float v) {
#if defined(__gfx1250__) || defined(__AMDGCN__)
  unsafeAtomicAdd(p, v);      // lowers to global_atomic_add_f32
#else
  atomicAdd(p, v);
#endif
}

// deg[n] = 1 (self loop)
__global__ void k_deg_init(float* __restrict__ deg, int n) {
  int i = blockIdx.x * blockDim.x + threadIdx.x;
  if (i < n) deg[i] = 1.0f;
}

// deg[col[e]] += 1
__global__ void k_deg_edges(const int* __restrict__ col, float* __restrict__ deg, int e) {
  int i = blockIdx.x * blockDim.x + threadIdx.x;
  if (i < e) atomAddF(&deg[col[i]], 1.0f);
}

// dinv[n] = rsqrt(deg[n])   (deg >= 1 always)
__global__ void k_dinv(const float* __restrict__ deg, float* __restrict__ dinv, int n) {
  int i = blockIdx.x * blockDim.x + threadIdx.x;
  if (i < n) dinv[i] = rsqrtf(deg[i]);
}

// norm[e] = dinv[row[e]] * dinv[col[e]]
__global__ void k_norm(const int* __restrict__ row, const int* __restrict__ col,
                       const float* __restrict__ dinv, float* __restrict__ norm, int e) {
  int i = blockIdx.x * blockDim.x + threadIdx.x;
  if (i < e) norm[i] = dinv[row[i]] * dinv[col[i]];
}

// init = relu(x0 @ W_in + b_in), via V_WMMA_F32_16X16X4_F32.
// Block = 128 threads = 4 waves; wave w computes output-col tile [16w, 16w+16).
// blockIdx.x = row tile (16 nodes). EXEC is all-ones through the WMMA loop.
__global__ void k_input_gemm(const float* __restrict__ x0,
                             const float* __restrict__ W_in,
                             const float* __restrict__ b_in,
                             float* __restrict__ init, int n) {
  const int wave = threadIdx.x >> 5;   // 0..3 -> N-tile
  const int lane = threadIdx.x & 31;
  const int half = lane >> 4;          // lanes 16-31 hold K+2 / M+8 halves
  const int mr   = lane & 15;
  const int m0   = blockIdx.x * 16;
  const int n0   = wave * 16;

  v8f acc = {};
  for (int kk = 0; kk < T_HIDDEN; kk += 4) {
    const int k0 = kk + half * 2;
    int mi = m0 + mr; if (mi >= n) mi = n - 1;   // clamp (no EXEC divergence)
    v2f a, b;
    // A 16x4 f32: vgpr v, lane l -> (M = l&15, K = (l>>4)*2 + v)
    a.x = x0[mi * T_HIDDEN + k0 + 0];
    a.y = x0[mi * T_HIDDEN + k0 + 1];
    // B 4x16 f32: vgpr v, lane l -> (K = (l>>4)*2 + v, Ncol = l&15)
    b.x = W_in[(k0 + 0) * T_D + n0 + mr];
    b.y = W_in[(k0 + 1) * T_D + n0 + mr];
    acc = __builtin_amdgcn_wmma_f32_16x16x4_f32(
        /*neg_a=*/false, a, /*neg_b=*/false, b,
        /*c_mod=*/(short)0, acc, /*reuse_a=*/false, /*reuse_b=*/false);
  }
  const float bias = b_in[n0 + mr];
#pragma unroll
  for (int v = 0; v < 8; ++v) {
    const int r = m0 + v + half * 8;   // C/D layout: vgpr v, half h -> M = v + 8h
    if (r < n) {
      float val = acc[v] + bias;
      init[r * T_D + n0 + mr] = val > 0.0f ? val : 0.0f;
    }
  }
}

// x[n, c*64+d] = track_h[c, n, d]   (c-major feature concat)
__global__ void k_tracks_init(const float* __restrict__ track_h,
                              float* __restrict__ x, int n) {
  long long idx = (long long)blockIdx.x * blockDim.x + threadIdx.x;
  if (idx < (long long)n * T_CD) {
    int node = (int)(idx >> 7);
    int j    = (int)(idx & (T_CD - 1));
    int c    = j >> 6;
    int d    = j & (T_D - 1);
    x[idx] = track_h[(long long)c * n * T_D + (long long)node * T_D + d];
  }
}

// Stage-1 diagonal: tmp = dinv^2 * x   (self-loop term of A@x)
__global__ void k_selfloop(const float* __restrict__ x, const float* __restrict__ dinv,
                           float* __restrict__ tmp, int n) {
  long long idx = (long long)blockIdx.x * blockDim.x + threadIdx.x;
  if (idx < (long long)n * T_CD) {
    int node = (int)(idx >> 7);
    float di = dinv[node];
    tmp[idx] = di * di * x[idx];
  }
}

// Stage-2 base: x = FW * dinv^2 * tmp + (1-FW) * init_x   (tiled init)
__global__ void k_base(const float* __restrict__ tmp, const float* __restrict__ dinv,
                       const float* __restrict__ init, float* __restrict__ x, int n) {
  long long idx = (long long)blockIdx.x * blockDim.x + threadIdx.x;
  if (idx < (long long)n * T_CD) {
    int node = (int)(idx >> 7);
    int j    = (int)(idx & (T_CD - 1));
    float di = dinv[node];
    x[idx] = T_FW * di * di * tmp[idx] + (1.0f - T_FW) * init[node * T_D + (j & (T_D - 1))];
  }
}

// Scatter SpMM: dst[col] += scale*norm * src[row]; one wave per edge,
// each lane owns 4 features (float4 gather burst = 512B/row, hits L2).
__global__ void k_edge_spmm(const float* __restrict__ src, float* __restrict__ dst,
                            const int* __restrict__ row, const int* __restrict__ col,
                            const float* __restrict__ norm, float scale, int e) {
  const int wv   = blockIdx.x * 8 + (threadIdx.x >> 5);  // 8 waves / block
  const int lane = threadIdx.x & 31;
  if (wv < e) {
    const int   r = row[wv];
    const int   c = col[wv];
    const float w = norm[wv] * scale;
    const float4 v = *(const float4*)(src + (long long)r * T_CD + lane * 4);
    float* p = dst + (long long)c * T_CD + lane * 4;
    atomAddF(p + 0, w * v.x);
    atomAddF(p + 1, w * v.y);
    atomAddF(p + 2, w * v.z);
    atomAddF(p + 3, w * v.w);
  }
}

// out[n,h] = W0*(a0*x[n,h]    + a1*x[n,64+h])
//          + W1*(a0*x[n,32+h] + a1*x[n,96+h]) + b
__global__ void k_final(const float* __restrict__ x, const float* __restrict__ attn,
                        const float* __restrict__ W_extra, const float* __restrict__ b_extra,
                        float* __restrict__ out, int n) {
  long long idx = (long long)blockIdx.x * blockDim.x + threadIdx.x;
  if (idx < (long long)n * T_HIDDEN) {
    int node = (int)(idx >> 5);
    int h    = (int)(idx & (T_HIDDEN - 1));
    const float a0 = attn[node * 2 + 0];
    const float a1 = attn[node * 2 + 1];
    const float* xr = x + (long long)node * T_CD;
    float e0 = a0 * xr[h]            + a1 * xr[T_D + h];
    float e1 = a0 * xr[T_HIDDEN + h] + a1 * xr[T_D + T_HIDDEN + h];
    out[idx] = e0 * W_extra[0] + e1 * W_extra[1] + b_extra[0];
  }
}

static inline size_t alignUp(size_t v) { return (v + 63) & ~(size_t)63; }

extern "C" void kernel_launch(void* const* d_in, const int* in_sizes, int n_in,
                              void* d_out, int out_size, void* d_ws, size_t ws_size,
                              hipStream_t stream) {
  const float* track_h = (const float*)d_in[0];   // [2, N, 64]
  const float* attn    = (const float*)d_in[1];   // [N, 2]
  const float* x0      = (const float*)d_in[2];   // [N, 32]
  const int*   eidx    = (const int*)  d_in[3];   // [2, E]
  const float* W_in    = (const float*)d_in[4];   // [32, 64]
  const float* b_in    = (const float*)d_in[5];   // [64]
  const float* W_ex    = (const float*)d_in[6];   // [2, 1]
  const float* b_ex    = (const float*)d_in[7];   // [1]
  float* out = (float*)d_out;

  const int N = in_sizes[2] / T_HIDDEN;
  const int E = in_sizes[3] / 2;
  const int* row = eidx;       // edge_index[0] = sources
  const int* col = eidx + E;   // edge_index[1] = targets

  // Workspace carve-up (floats), 256B-aligned slabs.
  float* ws = (float*)d_ws;
  size_t o = 0;
  float* deg  = ws + o; o += alignUp((size_t)N);
  float* dinv = ws + o; o += alignUp((size_t)N);
  float* norm = ws + o; o += alignUp((size_t)E);
  float* init = ws + o; o += alignUp((size_t)N * T_D);
  float* xb   = ws + o; o += alignUp((size_t)N * T_CD);
  float* tmp  = ws + o; o += alignUp((size_t)N * T_CD);
  (void)ws_size;

  const int BT = 256;
  const long long NF = (long long)N * T_CD;
  const int gN   = (N + BT - 1) / BT;
  const int gE   = (E + BT - 1) / BT;
  const int gNF  = (int)((NF + BT - 1) / BT);
  const int gEsp = (E + 7) / 8;                 // wave-per-edge, 8 waves/block
  const int gOut = (N * T_HIDDEN + BT - 1) / BT;

  // --- GCN norm build ---
  k_deg_init <<<gN, BT, 0, stream>>>(deg, N);
  k_deg_edges<<<gE, BT, 0, stream>>>(col, deg, E);
  k_dinv     <<<gN, BT, 0, stream>>>(deg, dinv, N);
  k_norm     <<<gE, BT, 0, stream>>>(row, col, dinv, norm, E);

  // --- input layer (WMMA f32 GEMM + ReLU) ---
  k_input_gemm<<<(N + 15) / 16, 128, 0, stream>>>(x0, W_in, b_in, init, N);

  // --- initial tracks layout [N, C*D] ---
  k_tracks_init<<<gNF, BT, 0, stream>>>(track_h, xb, N);

  // --- 4 layers: x = FW * A(Ax) + (1-FW) * init_x ---
  for (int l = 0; l < 4; ++l) {
    k_selfloop <<<gNF,  BT, 0, stream>>>(xb, dinv, tmp, N);
    k_edge_spmm<<<gEsp, BT, 0, stream>>>(xb, tmp, row, col, norm, 1.0f, E);
    k_base     <<<gNF,  BT, 0, stream>>>(tmp, dinv, init, xb, N);
    k_edge_spmm<<<gEsp, BT, 0, stream>>>(tmp, xb, row, col, norm, T_FW, E);
  }

  // --- attention-weighted class sum + head contraction ---
  k_final<<<gOut, BT, 0, stream>>>(xb, attn, W_ex, b_ex, out, N);
}